// L1Attn_79611513798903
// MI455X (gfx1250) — compile-verified
//
#include <hip/hip_runtime.h>
#include <cstdint>

#define NCTX 512
#define NH   8
#define WID  64
#define HPP  2                    // heads per LDS pass
#define ROWDW (HPP * WID + 4)     // 132 dwords: +4 pad -> stride%64=4, bank-conflict-free
#define TI   32                   // key tile
#define TJ   32                   // query tile

// Native clang vector: used ONLY for the final nontemporal stores (builtin requires it).
// The hot loop uses HIP float4 (struct) so SROA scalarizes it -> no register-tuple pressure.
typedef float f32x4 __attribute__((ext_vector_type(4)));

// CDNA5 async global->LDS copy, 16 bytes per lane, tracked by ASYNCcnt.
// GV mode: 64-bit global address in a VGPR pair, SADDR = off.
__device__ __forceinline__ void async_ld_b128(uint32_t lds_byte_off, const float* gaddr) {
    asm volatile("global_load_async_to_lds_b128 %0, %1, off"
                 :: "v"(lds_byte_off), "v"(gaddr)
                 : "memory");
}

__device__ __forceinline__ void wait_async0() {
    asm volatile("s_wait_asynccnt 0" ::: "memory");
}

__global__ __launch_bounds__(256) void l1attn_kernel(
    const float* __restrict__ q,   // [BS, NCTX(j), NH, WID]
    const float* __restrict__ k,   // [BS, NCTX(i), NH, WID]
    float* __restrict__ out)       // [BS, NCTX(i), NCTX(j), NH]
{
    __shared__ float ks[TI * ROWDW];
    __shared__ float qs[TJ * ROWDW];

    const int t  = threadIdx.x;     // 0..255
    const int jj = t & 15;          // query sub-index (coalesced store dimension)
    const int ii = t >> 4;          // key sub-index
    const int jB = blockIdx.x * TJ;
    const int iB = blockIdx.y * TI;
    const int b  = blockIdx.z;

    const float* qb = q + (size_t)b * NCTX * NH * WID;
    const float* kb = k + (size_t)b * NCTX * NH * WID;

    // flat-address low 32 bits of an LDS pointer == workgroup-relative LDS offset
    const uint32_t ksBase = (uint32_t)(uintptr_t)(&ks[0]);
    const uint32_t qsBase = (uint32_t)(uintptr_t)(&qs[0]);

    float acc[2][2][NH];   // [iPair][jPair][head], fully register-resident after unroll
#pragma unroll
    for (int a = 0; a < 2; ++a)
#pragma unroll
        for (int c = 0; c < 2; ++c)
#pragma unroll
            for (int h = 0; h < NH; ++h) acc[a][c][h] = 0.0f;

#pragma unroll
    for (int p = 0; p < NH / HPP; ++p) {   // 4 passes, heads {2p, 2p+1}
        // ---- cooperative async load: K tile + Q tile (1024 float4 each) ----
        // Per (b, row): HPP*WID = 128 contiguous floats starting at row*512 + p*128.
#pragma unroll
        for (int c = 0; c < 4; ++c) {
            const int f  = c * 256 + t;    // 0..1023
            const int r  = f >> 5;         // row 0..31
            const int c4 = f & 31;         // float4 index within row
            const uint32_t ldsOff = (uint32_t)((r * ROWDW + c4 * 4) * 4);
            const float* gK = kb + (size_t)(iB + r) * (NH * WID) + p * (HPP * WID) + c4 * 4;
            const float* gQ = qb + (size_t)(jB + r) * (NH * WID) + p * (HPP * WID) + c4 * 4;
            async_ld_b128(ksBase + ldsOff, gK);
            async_ld_b128(qsBase + ldsOff, gQ);
        }
        wait_async0();
        __syncthreads();

        // ---- compute: 2x2 register blocking, each ds_load_b128 feeds 2 outputs ----
#pragma unroll
        for (int hh = 0; hh < HPP; ++hh) {
            const int h = p * HPP + hh;
            float4 s00 = {0.f,0.f,0.f,0.f};  // (i=ii,    j=jj)
            float4 s01 = {0.f,0.f,0.f,0.f};  // (i=ii,    j=jj+16)
            float4 s10 = {0.f,0.f,0.f,0.f};  // (i=ii+16, j=jj)
            float4 s11 = {0.f,0.f,0.f,0.f};  // (i=ii+16, j=jj+16)
            const float* q0p = &qs[jj        * ROWDW + hh * WID];
            const float* q1p = &qs[(jj + 16) * ROWDW + hh * WID];
            const float* k0p = &ks[ii        * ROWDW + hh * WID];
            const float* k1p = &ks[(ii + 16) * ROWDW + hh * WID];
#pragma unroll
            for (int w = 0; w < WID; w += 4) {
                const float4 q0 = *(const float4*)(q0p + w);
                const float4 q1 = *(const float4*)(q1p + w);
                const float4 k0 = *(const float4*)(k0p + w);
                const float4 k1 = *(const float4*)(k1p + w);
                s00.x += fabsf(q0.x - k0.x); s00.y += fabsf(q0.y - k0.y);
                s00.z += fabsf(q0.z - k0.z); s00.w += fabsf(q0.w - k0.w);
                s01.x += fabsf(q1.x - k0.x); s01.y += fabsf(q1.y - k0.y);
                s01.z += fabsf(q1.z - k0.z); s01.w += fabsf(q1.w - k0.w);
                s10.x += fabsf(q0.x - k1.x); s10.y += fabsf(q0.y - k1.y);
                s10.z += fabsf(q0.z - k1.z); s10.w += fabsf(q0.w - k1.w);
                s11.x += fabsf(q1.x - k1.x); s11.y += fabsf(q1.y - k1.y);
                s11.z += fabsf(q1.z - k1.z); s11.w += fabsf(q1.w - k1.w);
            }
            acc[0][0][h] = s00.x + s00.y + s00.z + s00.w;
            acc[0][1][h] = s01.x + s01.y + s01.z + s01.w;
            acc[1][0][h] = s10.x + s10.y + s10.z + s10.w;
            acc[1][1][h] = s11.x + s11.y + s11.z + s11.w;
        }
        __syncthreads();   // before next pass overwrites LDS
    }

    // ---- store: out[((b*512 + i)*512 + j)*8 + h], h contiguous -> 2x b128 per pair, NT hint ----
    const float scale = -0.125f;   // -1/sqrt(64)
#pragma unroll
    for (int a = 0; a < 2; ++a) {
#pragma unroll
        for (int c = 0; c < 2; ++c) {
            const int i_ = iB + ii + a * 16;
            const int j_ = jB + jj + c * 16;
            const size_t off = (((size_t)b * NCTX + i_) * NCTX + j_) * NH;
            const f32x4 lo = { acc[a][c][0] * scale, acc[a][c][1] * scale,
                               acc[a][c][2] * scale, acc[a][c][3] * scale };
            const f32x4 hi = { acc[a][c][4] * scale, acc[a][c][5] * scale,
                               acc[a][c][6] * scale, acc[a][c][7] * scale };
            __builtin_nontemporal_store(lo, (f32x4*)(out + off));
            __builtin_nontemporal_store(hi, (f32x4*)(out + off + 4));
        }
    }
}

extern "C" void kernel_launch(void* const* d_in, const int* in_sizes, int n_in,
                              void* d_out, int out_size, void* d_ws, size_t ws_size,
                              hipStream_t stream) {
    const float* q = (const float*)d_in[0];
    const float* k = (const float*)d_in[1];
    float* out = (float*)d_out;
    const int bs = in_sizes[0] / (NCTX * NH * WID);   // = 2
    dim3 grid(NCTX / TJ, NCTX / TI, bs);              // (16, 16, 2)
    l1attn_kernel<<<grid, 256, 0, stream>>>(q, k, out);
}